// ComputeAllAtomCoords_29472065585617
// MI455X (gfx1250) — compile-verified
//
#include <hip/hip_runtime.h>
#include <math.h>

// ============================================================================
// MI455X (gfx1250) all-atom coordinate kernel.
//
// Roofline: ~17 MB read + ~129 MB write => memory bound (~6.3 us @ 23.3 TB/s).
// The 4x4 matrix chain is computed with V_WMMA_F32_16X16X4_F32, packing 4
// residues block-diagonally per 16x16 tile. The chain is kept TRANSPOSED so
// the running product is the B operand:   P_k^T = M_k^T * P_{k-1}^T
// => D(16x16) = A(16x4 = stacked M_k^T blocks) x B(4x16 = P^T blocks).
// Between steps, D's diagonal blocks are re-laid into B form with
// ds_bpermute_b32 (wave32 cross-lane), using the documented C/D layout.
// Per-class lookup tables are staged to LDS with CDNA5 async-to-LDS loads
// (GLOBAL_LOAD_ASYNC_TO_LDS_B128/B32, ASYNCcnt + s_wait_asynccnt).
//
// Layout assumptions (CDNA5 ISA 7.12.2):
//  A 16x4 f32 (2 VGPRs): lane m(<16): v0=A[m,K0], v1=A[m,K1];
//                        lane m+16 : v0=A[m,K2], v1=A[m,K3].   [documented]
//  C/D 16x16 f32 (8 VGPRs): vgpr r, lanes0-15 -> D[r,lane];
//                            lanes16-31 -> D[r+8,lane-16].     [documented]
//  B 4x16 f32 (2 VGPRs): v0: lanes0-15=B[K0,n], lanes16-31=B[K2,n];
//                        v1: K1 / K3.     [ASSUMED, mirrors A's K split]
// ============================================================================

typedef __attribute__((ext_vector_type(2))) float v2f;
typedef __attribute__((ext_vector_type(8))) float v8f;

#define WAVES_PER_BLOCK 8
#define RESIDUES_PER_TILE 4

__device__ inline float bperm_f(int byte_idx, float v) {
  return __int_as_float(__builtin_amdgcn_ds_bpermute(byte_idx, __float_as_int(v)));
}

// D = A x B + 0 via V_WMMA_F32_16X16X4_F32
__device__ inline v8f wmma4(v2f a, v2f b) {
  v8f c = {0.f, 0.f, 0.f, 0.f, 0.f, 0.f, 0.f, 0.f};
  return __builtin_amdgcn_wmma_f32_16x16x4_f32(
      /*neg_a=*/false, a, /*neg_b=*/false, b,
      /*c_mod=*/(short)0, c, /*reuse_a=*/false, /*reuse_b=*/false);
}

// CDNA5 async copy: global -> LDS, 16 bytes per enabled lane. VDST is the LDS
// byte address (low 32 bits of the flat shared address), VADDR 64-bit global.
__device__ inline void async_g2lds_b128(const void* gaddr, void* lds) {
  unsigned l = (unsigned)(size_t)lds;
  unsigned long long g = (unsigned long long)(size_t)gaddr;
  asm volatile("global_load_async_to_lds_b128 %0, %1, off" :: "v"(l), "v"(g) : "memory");
}
__device__ inline void async_g2lds_b32(const void* gaddr, void* lds) {
  unsigned l = (unsigned)(size_t)lds;
  unsigned long long g = (unsigned long long)(size_t)gaddr;
  asm volatile("global_load_async_to_lds_b32 %0, %1, off" :: "v"(l), "v"(g) : "memory");
}
__device__ inline void wait_async0() {
  asm volatile("s_wait_asynccnt 0x0" ::: "memory");
}

// Column i of M = RT @ rotX(c,s)  (branch-free binary-tree select):
//   col0 = RT[:,0]; col1 = c*RT[:,1]+s*RT[:,2]; col2 = -s*RT[:,1]+c*RT[:,2]; col3 = RT[:,3]
__device__ inline float melem(const float4& row, int i, float c, float s) {
  float m1 = c * row.y + s * row.z;
  float m2 = c * row.z - s * row.y;
  float lo = (i & 1) ? m1 : row.x;     // i==0 -> row.x, i==1 -> m1
  float hi = (i & 1) ? row.w : m2;     // i==2 -> m2,   i==3 -> row.w
  return (i & 2) ? hi : lo;
}

// Build A = M_k^T blocks (A[4g+i, j] = M_g[j, i]) directly in the A layout.
// lane m=4g+i (m=lane&15), jb=(lane>>4)*2: a.x = M_g[jb, i], a.y = M_g[jb+1, i]
// Normalization uses HW v_sqrt_f32 / v_rcp_f32 (<=1 ulp each; inputs are O(1)
// random-normal, no denormal/inf edge cases).
__device__ inline v2f build_A(const float* __restrict__ sRT,
                              const float* __restrict__ alphas,
                              int seqA, int rA, int lane, int k) {
  int i  = lane & 3;
  int jb = (lane >> 4) * 2;
  const float* rt = sRT + ((size_t)(seqA * 10 + k) * 4 + jb) * 4;
  float4 row0 = *(const float4*)rt;        // RT[jb, :]
  float4 row1 = *(const float4*)(rt + 4);  // RT[jb+1, :]
  float2 cs = *(const float2*)(alphas + ((size_t)rA * 10 + k) * 2);
  float n   = __builtin_amdgcn_sqrtf(cs.x * cs.x + cs.y * cs.y) + 1e-6f;
  float inv = __builtin_amdgcn_rcpf(n);
  float c = cs.x * inv, s = cs.y * inv;
  v2f a;
  a.x = melem(row0, i, c, s);
  a.y = melem(row1, i, c, s);
  return a;
}

// Re-layout D's diagonal blocks into B form for the next step:
// B[j, 4g+c] = newP_g^T[j, c] = D[4g+j, 4g+c].
// Source lane: 0-7 -> lane, 8-15 -> lane+16, 16-23 -> lane-16, 24-31 -> lane.
__device__ inline v2f conv_D_to_B(const v8f& d, int lane) {
  int src = lane + (((lane >> 3) & 1) << 4) - (((lane >> 4) & 1) << 4);
  int idx4 = src << 2;
  bool up  = lane >= 16;          // K rows {2,3} vs {0,1}
  bool odd = (lane >> 2) & 1;     // residue parity within half
  float t0 = bperm_f(idx4, d[0]);
  float t2 = bperm_f(idx4, d[2]);
  float t4 = bperm_f(idx4, d[4]);
  float t6 = bperm_f(idx4, d[6]);
  float t1 = bperm_f(idx4, d[1]);
  float t3 = bperm_f(idx4, d[3]);
  float t5 = bperm_f(idx4, d[5]);
  float t7 = bperm_f(idx4, d[7]);
  v2f b;
  b.x = odd ? (up ? t6 : t4) : (up ? t2 : t0);
  b.y = odd ? (up ? t7 : t5) : (up ? t3 : t1);
  return b;
}

// Emit frame f (D holds newP^T diagonal blocks). Active lanes (0-7, 24-31)
// each own one full output row P_g[j, 0..3] as 4 consecutive accumulator VGPRs
// at their own lane -> contiguous float4 store to global + LDS stash.
__device__ inline void emit_frame(const v8f& d, int lane, int r0, int f,
                                  float* __restrict__ outFrames,
                                  float* __restrict__ sFwave) {
  int grp3 = (lane >> 3) & 3;
  bool active = (grp3 == 0) || (grp3 == 3);
  int j = lane & 3;
  int g = (lane & 15) >> 2;
  bool hi = (lane >> 2) & 1;    // residue in {1,3} -> rows live in d[4..7]
  float4 row;
  row.x = hi ? d[4] : d[0];
  row.y = hi ? d[5] : d[1];
  row.z = hi ? d[6] : d[2];
  row.w = hi ? d[7] : d[3];
  if (active) {
    int r = r0 + g;
    *(float4*)(outFrames + ((size_t)r * 9 + f) * 16 + j * 4) = row;
    *(float4*)(sFwave + (size_t)(g * 9 + f) * 16 + j * 4)    = row;
  }
}

__global__ __launch_bounds__(256)
void allatom_wmma_kernel(const float* __restrict__ alphas,   // (L,10,2)
                         const float* __restrict__ Rs,       // (L,3,3)
                         const float* __restrict__ Ts,       // (L,3)
                         const float* __restrict__ RTbase,   // (5,10,4,4)
                         const float* __restrict__ xyzBase,  // (5,34,4)
                         const int*   __restrict__ baseIdx,  // (5,34)
                         const int*   __restrict__ seqIdx,   // (L,)
                         float* __restrict__ out,            // frames | xyz
                         int L) {
  __shared__ __align__(16) float sRT[5 * 10 * 16];   // 800 floats
  __shared__ __align__(16) float sXB[5 * 34 * 4];    // 680 floats
  __shared__ __align__(16) int sBI[5 * 34];          // 170 ints
  __shared__ __align__(16) float sF[WAVES_PER_BLOCK][4 * 9 * 16]; // frame stash

  int tid = threadIdx.x;
  // ---- stage tables to LDS via CDNA5 async-to-LDS DMA (ASYNCcnt) ----
  for (int i = tid; i < 200; i += 256)              // 800 floats as b128
    async_g2lds_b128(RTbase + i * 4, sRT + i * 4);
  for (int i = tid; i < 170; i += 256)              // 680 floats as b128
    async_g2lds_b128(xyzBase + i * 4, sXB + i * 4);
  for (int i = tid; i < 170; i += 256)              // 170 ints as b32
    async_g2lds_b32(baseIdx + i, sBI + i);
  wait_async0();
  __syncthreads();

  int lane = tid & 31;
  int wv   = tid >> 5;
  int tile = blockIdx.x * WAVES_PER_BLOCK + wv;
  int r0   = tile * RESIDUES_PER_TILE;
  bool tileOK = (r0 + RESIDUES_PER_TILE) <= L;   // wave-uniform

  if (tileOK) {
    int mg   = (lane & 15) >> 2;   // residue index within tile for A/B roles
    int rA   = r0 + mg;
    int seqA = seqIdx[rA];

    // ---- frame 0 (RTF0 from Rs/Ts), lanes 0..15 write one row each ----
    if (lane < 16) {
      int g = lane >> 2, c = lane & 3, r = r0 + g;
      int cl = (c < 3) ? c : 2;                      // clamped (safe loads)
      float4 v;
      v.x = Rs[(size_t)r * 9 + cl * 3 + 0];
      v.y = Rs[(size_t)r * 9 + cl * 3 + 1];
      v.z = Rs[(size_t)r * 9 + cl * 3 + 2];
      v.w = Ts[(size_t)r * 3 + cl];
      if (c == 3) { v.x = 0.f; v.y = 0.f; v.z = 0.f; v.w = 1.f; }
      *(float4*)(out + ((size_t)r * 9 + 0) * 16 + c * 4) = v;
      *(float4*)(&sF[wv][(size_t)(g * 9 + 0) * 16 + c * 4]) = v;
    }

    // ---- B0 = RTF0^T blocks: B[j, 4g+c] = RTF0_g[c, j] (branch-free) ----
    v2f B;
    {
      int c  = lane & 3;
      int jb = (lane >> 4) * 2;                      // {0,2}
      int cl = (c < 3) ? c : 2;                      // clamped (safe loads)
      float rs_j  = Rs[(size_t)rA * 9 + cl * 3 + jb];     // RTF0[c, jb],  c<3
      float rs_j1 = Rs[(size_t)rA * 9 + cl * 3 + 1];      // RTF0[c, 1],   c<3
      float ts    = Ts[(size_t)rA * 3 + cl];              // RTF0[c, 3],   c<3
      float vy    = (jb == 0) ? rs_j1 : ts;               // RTF0[c, jb+1]
      B.x = (c < 3) ? rs_j : 0.f;                         // row3 cols {0,2} = 0
      B.y = (c < 3) ? vy : ((jb == 0) ? 0.f : 1.f);       // row3: col1=0, col3=1
    }

    // ---- chain: frames 1..8, torsions 2..9, branch points at RTF3/RTF6 ----
    v2f A;
    v8f D;
    A = build_A(sRT, alphas, seqA, rA, lane, 2); D = wmma4(A, B);
    emit_frame(D, lane, r0, 1, out, sF[wv]); B = conv_D_to_B(D, lane);

    A = build_A(sRT, alphas, seqA, rA, lane, 3); D = wmma4(A, B);
    emit_frame(D, lane, r0, 2, out, sF[wv]); B = conv_D_to_B(D, lane);

    A = build_A(sRT, alphas, seqA, rA, lane, 4); D = wmma4(A, B);
    emit_frame(D, lane, r0, 3, out, sF[wv]);
    v2f B3 = conv_D_to_B(D, lane);                     // RTF3 (branch point)

    A = build_A(sRT, alphas, seqA, rA, lane, 5); D = wmma4(A, B3);
    emit_frame(D, lane, r0, 4, out, sF[wv]);           // RTF4 (terminal)

    A = build_A(sRT, alphas, seqA, rA, lane, 6); D = wmma4(A, B3);
    emit_frame(D, lane, r0, 5, out, sF[wv]); B = conv_D_to_B(D, lane); // RTF5

    A = build_A(sRT, alphas, seqA, rA, lane, 7); D = wmma4(A, B);
    emit_frame(D, lane, r0, 6, out, sF[wv]);
    v2f B6 = conv_D_to_B(D, lane);                     // RTF6 (branch point)

    A = build_A(sRT, alphas, seqA, rA, lane, 8); D = wmma4(A, B6);
    emit_frame(D, lane, r0, 7, out, sF[wv]);           // RTF7 (terminal)

    A = build_A(sRT, alphas, seqA, rA, lane, 9); D = wmma4(A, B6);
    emit_frame(D, lane, r0, 8, out, sF[wv]);           // RTF8 (terminal)
  }

  __syncthreads();   // uniform; sF stash complete for all waves

  if (tileOK) {
    // ---- atom phase: 4 residues x 34 atoms, 8 lanes per residue ----
    int g  = lane >> 3;
    int tl = lane & 7;
    int r  = r0 + g;
    int seq = seqIdx[r];
    float* xout = out + (size_t)L * 144;   // xyz region after frames
    for (int b = 0; b < 40; b += 8) {
      int t = b + tl;
      if (t < 34) {
        int bi = sBI[seq * 34 + t];
        float4 x = *(const float4*)&sXB[(size_t)(seq * 34 + t) * 4];
        const float* F = &sF[wv][(size_t)(g * 9 + bi) * 16];
        float o0 = F[0] * x.x + F[1] * x.y + F[2]  * x.z + F[3]  * x.w;
        float o1 = F[4] * x.x + F[5] * x.y + F[6]  * x.z + F[7]  * x.w;
        float o2 = F[8] * x.x + F[9] * x.y + F[10] * x.z + F[11] * x.w;
        size_t o = ((size_t)r * 34 + t) * 3;
        xout[o + 0] = o0; xout[o + 1] = o1; xout[o + 2] = o2;
      }
    }
  }
}

extern "C" void kernel_launch(void* const* d_in, const int* in_sizes, int n_in,
                              void* d_out, int out_size, void* d_ws, size_t ws_size,
                              hipStream_t stream) {
  const float* alphas  = (const float*)d_in[0];
  const float* Rs      = (const float*)d_in[1];
  const float* Ts      = (const float*)d_in[2];
  const float* RTbase  = (const float*)d_in[3];
  const float* xyzBase = (const float*)d_in[4];
  const int*   baseIdx = (const int*)d_in[5];
  const int*   seqIdx  = (const int*)d_in[6];
  int L = in_sizes[1] / 9;                 // Rs is (L,3,3)
  float* out = (float*)d_out;

  int residues_per_block = WAVES_PER_BLOCK * RESIDUES_PER_TILE;  // 32
  int blocks = (L + residues_per_block - 1) / residues_per_block;
  hipLaunchKernelGGL(allatom_wmma_kernel, dim3(blocks), dim3(256), 0, stream,
                     alphas, Rs, Ts, RTbase, xyzBase, baseIdx, seqIdx, out, L);
}